// MatMul_6502580486270
// MI455X (gfx1250) — compile-verified
//
#include <hip/hip_runtime.h>
#include <stdint.h>

// ---------------------------------------------------------------------------
// FP32 GEMM for MI455X (gfx1250) using V_WMMA_F32_16X16X4_F32 with
// double-buffered GLOBAL_LOAD_ASYNC_TO_LDS_B128 pipelining (ASYNCcnt).
// C[4096x4096] = A[4096x4096] * B[4096x4096], all fp32 row-major.
//
// Block: 256 threads (8 wave32) -> 128x128 C tile.
// Wave (wm in 0..3, wn in 0..1): 32x64 sub-tile = 2x4 grid of 16x16 WMMA accs.
// K blocked at 32; 8 WMMA k-steps of 4 per K-block.
// ---------------------------------------------------------------------------

typedef __attribute__((ext_vector_type(2))) float v2f;
typedef __attribute__((ext_vector_type(8))) float v8f;

#define MAT_M 4096
#define MAT_N 4096
#define MAT_K 4096

#define TILE_M 128
#define TILE_N 128
#define TILE_K 32

// LDS pitches (floats). Multiples of 4 (16B-aligned rows for b128 async
// copies) and bank-conflict-free for the WMMA fragment reads:
//  - LDA=36: column reads stride 36 dwords; {m*36 mod 64} distinct for
//    m=0..15; the +2-dword offset of lanes 16..31 is disjoint mod 4.
//  - LDB=144: 144 mod 64 == 16, so the row read by lanes 16..31 sits 16
//    banks above lanes 0..15 (fragment col base is a multiple of 16).
#define LDA 36
#define LDB 144

// Async copy of 16 bytes: global -> LDS, tracked by ASYNCcnt, no VGPR staging.
// lds_addr is the LDS byte address (low 32 bits of the generic pointer, per
// ISA section 10.2).
__device__ __forceinline__ void async_copy_b128(uint32_t lds_addr, uint64_t gaddr) {
    asm volatile("global_load_async_to_lds_b128 %0, %1, off"
                 :: "v"(lds_addr), "v"(gaddr) : "memory");
}

__device__ __forceinline__ void wait_async_zero() {
    asm volatile("s_wait_asynccnt 0" ::: "memory");
}

__global__ __launch_bounds__(256, 2)
void gemm_f32_wmma(const float* __restrict__ A,
                   const float* __restrict__ B,
                   float* __restrict__ C) {
    // Double-buffered tiles: 2 * (18 KB + 18 KB) = 72 KB of the 320 KB pool.
    __shared__ __align__(16) float As[2][TILE_M * LDA];
    __shared__ __align__(16) float Bs[2][TILE_K * LDB];

    const int tid  = threadIdx.x;             // 0..255
    const int wave = tid >> 5;                // 0..7
    const int lane = tid & 31;                // 0..31
    const int lh   = lane >> 4;               // lane half: 0 or 1
    const int lm   = lane & 15;               // lane-in-half: 0..15

    const int wm   = wave & 3;                // wave row: 0..3 (32 rows each)
    const int wn   = wave >> 2;               // wave col: 0..1 (64 cols each)

    const int bm   = blockIdx.y * TILE_M;     // C-tile row base
    const int bn   = blockIdx.x * TILE_N;     // C-tile col base

    // Accumulators: 2 (m-tiles) x 4 (n-tiles) of 16x16 f32.
    v8f acc[2][4];
#pragma unroll
    for (int i = 0; i < 2; ++i)
#pragma unroll
        for (int j = 0; j < 4; ++j)
            acc[i][j] = (v8f)(0.0f);

    const int nkb = MAT_K / TILE_K;           // 128 K-blocks

    // Issue the async tile copies for K-block kb into buffer `buf`.
    auto issue_tile = [&](int kb, int buf) {
        const int k0g = kb * TILE_K;
        const uint32_t aBase = (uint32_t)(uintptr_t)&As[buf][0];
        const uint32_t bBase = (uint32_t)(uintptr_t)&Bs[buf][0];
        // A tile: 128 rows x 32 cols = 1024 b128 copies, coalesced.
#pragma unroll
        for (int i = 0; i < 4; ++i) {
            const int idx = i * 256 + tid;    // 0..1023
            const int row = idx >> 3;         // 8 float4 per row
            const int c4  = idx & 7;
            const uint64_t g = (uint64_t)(uintptr_t)
                (A + (size_t)(bm + row) * MAT_K + k0g + c4 * 4);
            async_copy_b128(aBase + (uint32_t)(row * LDA + c4 * 4) * 4u, g);
        }
        // B tile: 32 rows x 128 cols = 1024 b128 copies, coalesced.
#pragma unroll
        for (int i = 0; i < 4; ++i) {
            const int idx = i * 256 + tid;    // 0..1023
            const int row = idx >> 5;         // 32 float4 per row
            const int c4  = idx & 31;
            const uint64_t g = (uint64_t)(uintptr_t)
                (B + (size_t)(k0g + row) * MAT_N + bn + c4 * 4);
            async_copy_b128(bBase + (uint32_t)(row * LDB + c4 * 4) * 4u, g);
        }
    };

    // Prologue: fill buffer 0.
    issue_tile(0, 0);
    wait_async_zero();
    __syncthreads();

    int buf = 0;
    for (int kb = 0; kb < nkb; ++kb) {
        // Stream the next K-panel into the other buffer while we compute.
        if (kb + 1 < nkb)
            issue_tile(kb + 1, buf ^ 1);

        const float* __restrict__ as = &As[buf][0];
        const float* __restrict__ bs = &Bs[buf][0];

        // ---- Compute: 8 k-steps of 4, 8 WMMAs each ----
#pragma unroll
        for (int kk = 0; kk < 8; ++kk) {
            const int k0 = kk * 4;

            // A fragments (16x4 f32): lanes 0-15 -> K = k0,k0+1;
            //                         lanes 16-31 -> K = k0+2,k0+3.
            v2f afrag[2];
#pragma unroll
            for (int tm = 0; tm < 2; ++tm) {
                const int r = wm * 32 + tm * 16 + lm;
                afrag[tm].x = as[r * LDA + k0 + 2 * lh + 0];
                afrag[tm].y = as[r * LDA + k0 + 2 * lh + 1];
            }

            // B fragments (4x16 f32): v0 holds rows k0 (lanes 0-15) / k0+1
            // (lanes 16-31); v1 holds rows k0+2 / k0+3.
            v2f bfrag[4];
#pragma unroll
            for (int tn = 0; tn < 4; ++tn) {
                const int c = wn * 64 + tn * 16 + lm;
                bfrag[tn].x = bs[(k0 + lh)     * LDB + c];
                bfrag[tn].y = bs[(k0 + 2 + lh) * LDB + c];
            }

#pragma unroll
            for (int tm = 0; tm < 2; ++tm)
#pragma unroll
                for (int tn = 0; tn < 4; ++tn)
                    acc[tm][tn] = __builtin_amdgcn_wmma_f32_16x16x4_f32(
                        /*neg_a=*/false, afrag[tm],
                        /*neg_b=*/false, bfrag[tn],
                        /*c_mod=*/(short)0, acc[tm][tn],
                        /*reuse_a=*/false, /*reuse_b=*/false);
        }

        // My async copies into buf^1 must have landed, and every wave must be
        // done reading buf, before the roles swap.
        wait_async_zero();
        __syncthreads();
        buf ^= 1;
    }

    // ---- Store C: VGPR r of a 16x16 f32 acc holds rows {r, r+8};
    //      lanes 0-15 cols n0..n0+15, lanes 16-31 same cols, rows +8. ----
#pragma unroll
    for (int tm = 0; tm < 2; ++tm) {
#pragma unroll
        for (int tn = 0; tn < 4; ++tn) {
            const int rbase = bm + wm * 32 + tm * 16 + lh * 8;
            const int col   = bn + wn * 64 + tn * 16 + lm;
#pragma unroll
            for (int r = 0; r < 8; ++r) {
                C[(size_t)(rbase + r) * MAT_N + col] = acc[tm][tn][r];
            }
        }
    }
}

extern "C" void kernel_launch(void* const* d_in, const int* in_sizes, int n_in,
                              void* d_out, int out_size, void* d_ws, size_t ws_size,
                              hipStream_t stream) {
    const float* A = (const float*)d_in[0];   // x: (4096, 4096) fp32 row-major
    const float* B = (const float*)d_in[1];   // y: (4096, 4096) fp32 row-major
    float* C = (float*)d_out;                 // (4096, 4096) fp32

    dim3 grid(MAT_N / TILE_N, MAT_M / TILE_M);   // (32, 32)
    dim3 block(256);
    gemm_f32_wmma<<<grid, block, 0, stream>>>(A, B, C);
}